// SS1D_MB_23192823398801
// MI455X (gfx1250) — compile-verified
//
#include <hip/hip_runtime.h>
#include <hip/hip_bf16.h>
#include <math.h>

// ---------------- problem constants ----------------
constexpr int BSZ   = 2;
constexpr int SEQ   = 2048;
constexpr int DM    = 768;          // d_model
constexpr int DI    = 1536;         // d_inner
constexpr int DSTATE= 16;
constexpr int DTR   = 48;           // dt_rank
constexpr int R2N   = DTR + 2*DSTATE; // 80
constexpr int MROWS = BSZ*SEQ;      // 4096 flattened rows
constexpr float EPS = 1e-5f;

// chunked scan geometry
constexpr int NCH   = 32;           // chunks per (b,dir)
constexpr int CHT   = SEQ / NCH;    // 64 steps per chunk

// ---------------- WMMA f32 16x16x4 helpers ----------------
typedef float v2f __attribute__((ext_vector_type(2)));
typedef float v8f __attribute__((ext_vector_type(8)));

__device__ __forceinline__ v8f wmma_f32(v2f a, v2f b, v8f c) {
  // 8 args: (neg_a, A, neg_b, B, c_mod, C, reuse_a, reuse_b)
  return __builtin_amdgcn_wmma_f32_16x16x4_f32(false, a, false, b, (short)0, c, false, false);
}

// A fragment: 16(M) x 4(K) f32.  Lane l: m = l&15 ; lanes 0-15 hold K={k,k+1},
// lanes 16-31 hold K={k+2,k+3}  (ISA 7.12.2, 32-bit A 16x4 layout).
__device__ __forceinline__ v2f load_a_frag(const float* __restrict__ A, int lda,
                                           int row0, int k, int lane) {
  int m  = row0 + (lane & 15);
  int kk = k + ((lane >> 4) << 1);
  const float* p = A + (size_t)m * lda + kk;
  v2f a; a.x = p[0]; a.y = p[1];
  return a;
}
// B fragment: 4(K) x 16(N) f32, mirrored layout.
__device__ __forceinline__ v2f load_b_frag(const float* __restrict__ B, int ldb,
                                           int k, int col0, int lane) {
  int n  = col0 + (lane & 15);
  int kk = k + ((lane >> 4) << 1);
  v2f b; b.x = B[(size_t)kk * ldb + n]; b.y = B[(size_t)(kk + 1) * ldb + n];
  return b;
}

__device__ __forceinline__ float silu(float x) { return x / (1.f + expf(-x)); }
__device__ __forceinline__ float softplus(float x) {
  return (x > 20.f) ? x : log1pf(expf(x));
}

// ---------------- async global->LDS copy (CDNA5), guarded ----------------
#if defined(__has_builtin)
#if __has_builtin(__builtin_amdgcn_global_load_async_to_lds_b32) && \
    __has_builtin(__builtin_amdgcn_s_wait_asynccnt)
#define ASYNC_LDS 1
#endif
#endif
#ifndef ASYNC_LDS
#define ASYNC_LDS 0
#endif

__device__ __forceinline__ void async_copy_f32(const float* gsrc, float* lds_dst) {
#if ASYNC_LDS
  __builtin_amdgcn_global_load_async_to_lds_b32(
      (__attribute__((address_space(1))) int*)gsrc,
      (__attribute__((address_space(3))) int*)lds_dst, 0, 0);
#else
  *lds_dst = *gsrc;
#endif
}
__device__ __forceinline__ void async_wait() {
#if ASYNC_LDS
  __builtin_amdgcn_s_wait_asynccnt(0);
#endif
}

// =====================================================================
// K1: xn = layernorm(x, gamma_in, beta_in)       rows of 768, block=256
// =====================================================================
__global__ void k1_ln_in(const float* __restrict__ x, const float* __restrict__ g,
                         const float* __restrict__ b, float* __restrict__ xn) {
  const int row = blockIdx.x;
  const int tid = threadIdx.x;
  const float* xr = x + (size_t)row * DM;
  __shared__ float red[256];

  float v[3]; float s = 0.f;
#pragma unroll
  for (int i = 0; i < 3; ++i) { v[i] = xr[tid + i * 256]; s += v[i]; }
  red[tid] = s; __syncthreads();
  for (int o = 128; o > 0; o >>= 1) { if (tid < o) red[tid] += red[tid + o]; __syncthreads(); }
  const float mu = red[0] * (1.f / DM);
  __syncthreads();

  float vs = 0.f;
#pragma unroll
  for (int i = 0; i < 3; ++i) { float d = v[i] - mu; vs += d * d; }
  red[tid] = vs; __syncthreads();
  for (int o = 128; o > 0; o >>= 1) { if (tid < o) red[tid] += red[tid + o]; __syncthreads(); }
  const float rstd = rsqrtf(red[0] * (1.f / DM) + EPS);

#pragma unroll
  for (int i = 0; i < 3; ++i) {
    int c = tid + i * 256;
    xn[(size_t)row * DM + c] = (v[i] - mu) * rstd * g[c] + b[c];
  }
}

// =====================================================================
// K2: xz = xn @ W_in  (4096x768 @ 768x3072), split into xi | z
//     wave tile 32x64 (8 accumulators), K-step 4.  waves = 128*48
// =====================================================================
__global__ void k2_gemm_in(const float* __restrict__ A, const float* __restrict__ W,
                           float* __restrict__ xi, float* __restrict__ z) {
  const int lane = threadIdx.x & 31;
  const int wave = (int)((blockIdx.x * blockDim.x + threadIdx.x) >> 5);
  constexpr int NT = (2 * DI) / 64;               // 48
  if (wave >= (MROWS / 32) * NT) return;
  const int row0 = (wave / NT) * 32;
  const int col0 = (wave % NT) * 64;

  v8f acc[2][4] = {};
  for (int k = 0; k < DM; k += 4) {
    v2f a0 = load_a_frag(A, DM, row0,      k, lane);
    v2f a1 = load_a_frag(A, DM, row0 + 16, k, lane);
    __builtin_prefetch(A + (size_t)(row0 + (lane & 15)) * DM + k + 16, 0, 3);
#pragma unroll
    for (int j = 0; j < 4; ++j) {
      v2f b = load_b_frag(W, 2 * DI, k, col0 + j * 16, lane);
      acc[0][j] = wmma_f32(a0, b, acc[0][j]);
      acc[1][j] = wmma_f32(a1, b, acc[1][j]);
    }
  }
#pragma unroll
  for (int h = 0; h < 2; ++h) {
#pragma unroll
    for (int j = 0; j < 4; ++j) {
      int n = col0 + j * 16 + (lane & 15);
      float* dst = (n < DI) ? xi : z;
      int nn = (n < DI) ? n : n - DI;
#pragma unroll
      for (int r = 0; r < 8; ++r) {
        int m = row0 + h * 16 + ((lane >> 4) << 3) + r;
        dst[(size_t)m * DI + nn] = acc[h][j][r];
      }
    }
  }
}

// =====================================================================
// K3: u = silu(depthwise_conv1d(xi, conv_w, conv_b))   K=3, pad=1
// =====================================================================
__global__ void k3_conv_silu(const float* __restrict__ xi, const float* __restrict__ cw,
                             const float* __restrict__ cb, float* __restrict__ u) {
  const long long idx = (long long)blockIdx.x * blockDim.x + threadIdx.x;
  const long long total = (long long)MROWS * DI;
  if (idx >= total) return;
  const int d = (int)(idx % DI);
  const int t = (int)((idx / DI) % SEQ);
  const int b = (int)(idx / ((long long)DI * SEQ));
  float acc = cb[d];
#pragma unroll
  for (int k = 0; k < 3; ++k) {
    int tt = t + k - 1;
    if (tt >= 0 && tt < SEQ)
      acc += xi[((size_t)b * SEQ + tt) * DI + d] * cw[d * 3 + k];
  }
  u[idx] = silu(acc);
}

// =====================================================================
// K4: proj[dir] = u @ x_proj_w[dir]  (4096x1536 @ 1536x80), wave tile 16x16
// =====================================================================
__global__ void k4_gemm_proj(const float* __restrict__ u, const float* __restrict__ xpw,
                             float* __restrict__ proj) {
  const int lane = threadIdx.x & 31;
  int wave = (int)((blockIdx.x * blockDim.x + threadIdx.x) >> 5);
  constexpr int NT = R2N / 16;                    // 5
  constexpr int TPD = (MROWS / 16) * NT;
  if (wave >= 2 * TPD) return;
  const int dir = wave / TPD;  wave %= TPD;
  const int row0 = (wave / NT) * 16;
  const int col0 = (wave % NT) * 16;
  const float* W = xpw + (size_t)dir * DI * R2N;

  v8f acc = {};
  for (int k = 0; k < DI; k += 4) {
    v2f a = load_a_frag(u, DI, row0, k, lane);
    v2f b = load_b_frag(W, R2N, k, col0, lane);
    acc = wmma_f32(a, b, acc);
  }
  float* pd = proj + (size_t)dir * MROWS * R2N;
  const int n = col0 + (lane & 15);
#pragma unroll
  for (int r = 0; r < 8; ++r) {
    int m = row0 + ((lane >> 4) << 3) + r;
    pd[(size_t)m * R2N + n] = acc[r];
  }
}

// =====================================================================
// K5: dt[dir] = softplus(proj[dir][:, :48] @ dt_w[dir] + dt_b[dir])
//     (4096x48 @ 48x1536), wave tile 16x64
// =====================================================================
__global__ void k5_gemm_dt(const float* __restrict__ proj, const float* __restrict__ dtw,
                           const float* __restrict__ dtb, float* __restrict__ dt) {
  const int lane = threadIdx.x & 31;
  int wave = (int)((blockIdx.x * blockDim.x + threadIdx.x) >> 5);
  constexpr int NT = DI / 64;                     // 24
  constexpr int TPD = (MROWS / 16) * NT;
  if (wave >= 2 * TPD) return;
  const int dir = wave / TPD;  wave %= TPD;
  const int row0 = (wave / NT) * 16;
  const int col0 = (wave % NT) * 64;
  const float* Ap = proj + (size_t)dir * MROWS * R2N;   // lda = 80, cols 0..47
  const float* W  = dtw  + (size_t)dir * DTR * DI;
  const float* bias = dtb + (size_t)dir * DI;

  v8f acc[4] = {};
  for (int k = 0; k < DTR; k += 4) {
    v2f a = load_a_frag(Ap, R2N, row0, k, lane);
#pragma unroll
    for (int j = 0; j < 4; ++j) {
      v2f b = load_b_frag(W, DI, k, col0 + j * 16, lane);
      acc[j] = wmma_f32(a, b, acc[j]);
    }
  }
  float* dd = dt + (size_t)dir * MROWS * DI;
#pragma unroll
  for (int j = 0; j < 4; ++j) {
    int n = col0 + j * 16 + (lane & 15);
    float bn = bias[n];
#pragma unroll
    for (int r = 0; r < 8; ++r) {
      int m = row0 + ((lane >> 4) << 3) + r;
      dd[(size_t)m * DI + n] = softplus(acc[j][r] + bn);
    }
  }
}

// =====================================================================
// K6: chunked selective scan (3 passes).
//   pass A: per-chunk local scan from h=0 + running product of dA
//   pass B: sequential combine over chunks -> incoming state per chunk
//   pass C: re-scan each chunk from its incoming state, emit y
// grid for A/C: (DI/256, NCH, 2*BSZ), block=256. dir = z/BSZ, b = z%BSZ.
// =====================================================================
__global__ void k6a_scan_local(const float* __restrict__ u, const float* __restrict__ proj,
                               const float* __restrict__ dt, const float* __restrict__ A_log,
                               float* __restrict__ hbuf, float* __restrict__ pbuf) {
  const int tid = threadIdx.x;
  const int d   = blockIdx.x * 256 + tid;
  const int c   = blockIdx.y;
  const int dirb= blockIdx.z;               // dir*BSZ + b
  const int dir = dirb / BSZ;
  const int b   = dirb % BSZ;

  float A[DSTATE];
#pragma unroll
  for (int n = 0; n < DSTATE; ++n)
    A[n] = -expf(A_log[((size_t)dir * DI + d) * DSTATE + n]);

  float h[DSTATE] = {};
  float pr[DSTATE];
#pragma unroll
  for (int n = 0; n < DSTATE; ++n) pr[n] = 1.f;

  __shared__ float Bs[DSTATE];
  const float* pd = proj + (size_t)dir * MROWS * R2N;
  const float* dd = dt   + (size_t)dir * MROWS * DI;

  for (int s = 0; s < CHT; ++s) {
    const int g = c * CHT + s;
    const int t = dir ? (SEQ - 1 - g) : g;
    const size_t row = (size_t)b * SEQ + t;
    __syncthreads();
    if (tid < DSTATE) {
      async_copy_f32(pd + row * R2N + DTR + tid, &Bs[tid]);
      async_wait();
    }
    __syncthreads();

    const float dtv = dd[row * DI + d];
    const float uv  = u[row * DI + d];
    const float du  = dtv * uv;
#pragma unroll
    for (int n = 0; n < DSTATE; ++n) {
      float dA = expf(dtv * A[n]);
      h[n]  = dA * h[n] + du * Bs[n];
      pr[n] *= dA;
    }
  }
  const size_t base = (((size_t)dirb * NCH + c) * DI + d) * DSTATE;
#pragma unroll
  for (int n = 0; n < DSTATE; ++n) { hbuf[base + n] = h[n]; pbuf[base + n] = pr[n]; }
}

__global__ void k6b_scan_combine(float* __restrict__ hbuf, const float* __restrict__ pbuf) {
  const int gid = blockIdx.x * blockDim.x + threadIdx.x;   // < 2*BSZ*DI*DSTATE
  const int dirb = gid / (DI * DSTATE);
  const int rem  = gid % (DI * DSTATE);
  float h = 0.f;
  for (int c = 0; c < NCH; ++c) {
    const size_t idx = ((size_t)dirb * NCH + c) * (size_t)(DI * DSTATE) + rem;
    const float hl = hbuf[idx];
    const float p  = pbuf[idx];
    hbuf[idx] = h;                 // becomes incoming state for chunk c
    h = p * h + hl;
  }
}

__global__ void k6c_scan_emit(const float* __restrict__ u, const float* __restrict__ proj,
                              const float* __restrict__ dt, const float* __restrict__ A_log,
                              const float* __restrict__ D_skip, const float* __restrict__ hbuf,
                              float* __restrict__ y2) {
  const int tid = threadIdx.x;
  const int d   = blockIdx.x * 256 + tid;
  const int c   = blockIdx.y;
  const int dirb= blockIdx.z;
  const int dir = dirb / BSZ;
  const int b   = dirb % BSZ;

  float A[DSTATE];
#pragma unroll
  for (int n = 0; n < DSTATE; ++n)
    A[n] = -expf(A_log[((size_t)dir * DI + d) * DSTATE + n]);
  const float dsk = D_skip[(size_t)dir * DI + d];

  float h[DSTATE];
  const size_t base = (((size_t)dirb * NCH + c) * DI + d) * DSTATE;
#pragma unroll
  for (int n = 0; n < DSTATE; ++n) h[n] = hbuf[base + n];

  __shared__ float BC[2 * DSTATE];          // B | C contiguous in proj row
  const float* pd = proj + (size_t)dir * MROWS * R2N;
  const float* dd = dt   + (size_t)dir * MROWS * DI;
  float*       yd = y2   + (size_t)dir * MROWS * DI;

  for (int s = 0; s < CHT; ++s) {
    const int g = c * CHT + s;
    const int t = dir ? (SEQ - 1 - g) : g;
    const size_t row = (size_t)b * SEQ + t;
    __syncthreads();
    if (tid < 2 * DSTATE) {
      async_copy_f32(pd + row * R2N + DTR + tid, &BC[tid]);
      async_wait();
    }
    __syncthreads();

    const float dtv = dd[row * DI + d];
    const float uv  = u[row * DI + d];
    const float du  = dtv * uv;
    float y = 0.f;
#pragma unroll
    for (int n = 0; n < DSTATE; ++n) {
      float dA = expf(dtv * A[n]);
      h[n] = dA * h[n] + du * BC[n];
      y   += h[n] * BC[DSTATE + n];
    }
    yd[row * DI + d] = y + uv * dsk;
  }
}

// =====================================================================
// K7: ym = w0*y_f + w1*y_b ; yg = (layernorm(ym)*g+b) * silu(z)
// =====================================================================
__global__ void k7_merge_ln_gate(const float* __restrict__ y2, const float* __restrict__ z,
                                 const float* __restrict__ mw, const float* __restrict__ g,
                                 const float* __restrict__ b, float* __restrict__ yg) {
  const int row = blockIdx.x;
  const int tid = threadIdx.x;
  const float w0 = mw[0], w1 = mw[1];
  const float* yf = y2 + (size_t)row * DI;
  const float* yb = y2 + (size_t)MROWS * DI + (size_t)row * DI;
  __shared__ float red[256];

  float v[6]; float s = 0.f;
#pragma unroll
  for (int i = 0; i < 6; ++i) {
    int c = tid + i * 256;
    v[i] = w0 * yf[c] + w1 * yb[c];
    s += v[i];
  }
  red[tid] = s; __syncthreads();
  for (int o = 128; o > 0; o >>= 1) { if (tid < o) red[tid] += red[tid + o]; __syncthreads(); }
  const float mu = red[0] * (1.f / DI);
  __syncthreads();

  float vs = 0.f;
#pragma unroll
  for (int i = 0; i < 6; ++i) { float d = v[i] - mu; vs += d * d; }
  red[tid] = vs; __syncthreads();
  for (int o = 128; o > 0; o >>= 1) { if (tid < o) red[tid] += red[tid + o]; __syncthreads(); }
  const float rstd = rsqrtf(red[0] * (1.f / DI) + EPS);

#pragma unroll
  for (int i = 0; i < 6; ++i) {
    int c = tid + i * 256;
    float zc = z[(size_t)row * DI + c];
    yg[(size_t)row * DI + c] = ((v[i] - mu) * rstd * g[c] + b[c]) * silu(zc);
  }
}

// =====================================================================
// K8: out = yg @ W_out + x   (4096x1536 @ 1536x768), wave tile 32x64
// =====================================================================
__global__ void k8_gemm_out(const float* __restrict__ A, const float* __restrict__ W,
                            const float* __restrict__ x, float* __restrict__ out) {
  const int lane = threadIdx.x & 31;
  const int wave = (int)((blockIdx.x * blockDim.x + threadIdx.x) >> 5);
  constexpr int NT = DM / 64;                     // 12
  if (wave >= (MROWS / 32) * NT) return;
  const int row0 = (wave / NT) * 32;
  const int col0 = (wave % NT) * 64;

  v8f acc[2][4] = {};
  for (int k = 0; k < DI; k += 4) {
    v2f a0 = load_a_frag(A, DI, row0,      k, lane);
    v2f a1 = load_a_frag(A, DI, row0 + 16, k, lane);
    __builtin_prefetch(A + (size_t)(row0 + (lane & 15)) * DI + k + 16, 0, 3);
#pragma unroll
    for (int j = 0; j < 4; ++j) {
      v2f b = load_b_frag(W, DM, k, col0 + j * 16, lane);
      acc[0][j] = wmma_f32(a0, b, acc[0][j]);
      acc[1][j] = wmma_f32(a1, b, acc[1][j]);
    }
  }
#pragma unroll
  for (int h = 0; h < 2; ++h) {
#pragma unroll
    for (int j = 0; j < 4; ++j) {
      int n = col0 + j * 16 + (lane & 15);
#pragma unroll
      for (int r = 0; r < 8; ++r) {
        int m = row0 + h * 16 + ((lane >> 4) << 3) + r;
        out[(size_t)m * DM + n] = acc[h][j][r] + x[(size_t)m * DM + n];
      }
    }
  }
}

// =====================================================================
// launcher
// =====================================================================
extern "C" void kernel_launch(void* const* d_in, const int* in_sizes, int n_in,
                              void* d_out, int out_size, void* d_ws, size_t ws_size,
                              hipStream_t stream) {
  const float* x        = (const float*)d_in[0];
  const float* gamma_in = (const float*)d_in[1];
  const float* beta_in  = (const float*)d_in[2];
  const float* W_in     = (const float*)d_in[3];
  const float* conv_w   = (const float*)d_in[4];
  const float* conv_b   = (const float*)d_in[5];
  const float* x_proj_w = (const float*)d_in[6];
  const float* dt_w     = (const float*)d_in[7];
  const float* dt_b     = (const float*)d_in[8];
  const float* A_log    = (const float*)d_in[9];
  const float* D_skip   = (const float*)d_in[10];
  const float* merge_w  = (const float*)d_in[11];
  const float* gamma_out= (const float*)d_in[12];
  const float* beta_out = (const float*)d_in[13];
  const float* W_out    = (const float*)d_in[14];
  float* out = (float*)d_out;

  // workspace layout (fp32)
  float* ws   = (float*)d_ws;
  float* xn   = ws;                                   // 4096*768
  float* xi   = xn   + (size_t)MROWS * DM;            // 4096*1536
  float* z    = xi   + (size_t)MROWS * DI;            // 4096*1536
  float* u    = z    + (size_t)MROWS * DI;            // 4096*1536
  float* proj = u    + (size_t)MROWS * DI;            // 2*4096*80
  float* dt   = proj + (size_t)2 * MROWS * R2N;       // 2*4096*1536
  float* y2   = dt   + (size_t)2 * MROWS * DI;        // 2*4096*1536
  float* yg   = y2   + (size_t)2 * MROWS * DI;        // 4096*1536
  float* hbuf = yg   + (size_t)MROWS * DI;            // 4*NCH*DI*16
  float* pbuf = hbuf + (size_t)2 * BSZ * NCH * DI * DSTATE;

  // K1: layernorm_in
  k1_ln_in<<<MROWS, 256, 0, stream>>>(x, gamma_in, beta_in, xn);

  // K2: xz = xn @ W_in -> xi | z   (waves = 128*48 = 6144, 4 waves/block)
  k2_gemm_in<<<(MROWS / 32) * ((2 * DI) / 64) / 4, 128, 0, stream>>>(xn, W_in, xi, z);

  // K3: depthwise conv + silu
  {
    long long total = (long long)MROWS * DI;
    int blocks = (int)((total + 255) / 256);
    k3_conv_silu<<<blocks, 256, 0, stream>>>(xi, conv_w, conv_b, u);
  }

  // K4: proj per dir (waves = 2*256*5 = 2560)
  k4_gemm_proj<<<2 * (MROWS / 16) * (R2N / 16) / 4, 128, 0, stream>>>(u, x_proj_w, proj);

  // K5: dt per dir (waves = 2*256*24 = 12288)
  k5_gemm_dt<<<2 * (MROWS / 16) * (DI / 64) / 4, 128, 0, stream>>>(proj, dt_w, dt_b, dt);

  // K6: chunked scan (local -> combine -> emit)
  {
    dim3 grid(DI / 256, NCH, 2 * BSZ);
    k6a_scan_local<<<grid, 256, 0, stream>>>(u, proj, dt, A_log, hbuf, pbuf);
    int nthr = 2 * BSZ * DI * DSTATE;               // 98304
    k6b_scan_combine<<<nthr / 256, 256, 0, stream>>>(hbuf, pbuf);
    k6c_scan_emit<<<grid, 256, 0, stream>>>(u, proj, dt, A_log, D_skip, hbuf, y2);
  }

  // K7: merge + layernorm_out + gate
  k7_merge_ln_gate<<<MROWS, 256, 0, stream>>>(y2, z, merge_w, gamma_out, beta_out, yg);

  // K8: out = yg @ W_out + residual (waves = 128*12 = 1536)
  k8_gemm_out<<<(MROWS / 32) * (DM / 64) / 4, 128, 0, stream>>>(yg, W_out, x, out);
}